// SpikeRateReadout_30580167147913
// MI455X (gfx1250) — compile-verified
//
#include <hip/hip_runtime.h>
#include <math.h>

// ---------------------------------------------------------------------------
// SpikeRateReadout for MI455X (gfx1250)
//
// Stage 1 (bandwidth-bound, ~524 MB stream @ 23.3 TB/s => ~22.5us floor):
//   partial[c][b][n] = sum_{t in chunk c} spikes[b,t,n] * decay[t]
//   - float4 non-temporal loads (stream > 192MB L2, read-once)
//   - 4 T-chunks x 1024 waves = 4096 waves, t-loop unrolled x10 for MLP
// Stage 2 (tiny GEMM, uses V_WMMA_F32_16X16X4_F32 for full fp32 precision):
//   out[64,35] = (sum_c partial[c]) @ W^T + bias
//   - one wave per 16x16 tile, K=2048 in steps of 4
// ---------------------------------------------------------------------------

typedef __attribute__((ext_vector_type(2))) float v2f;
typedef __attribute__((ext_vector_type(4))) float v4f;
typedef __attribute__((ext_vector_type(8))) float v8f;

#define B_DIM   64
#define T_DIM   1000
#define N_DIM   2048
#define O_DIM   35
#define CHUNKS  4
#define TCHUNK  (T_DIM / CHUNKS)      // 250
#define N4      (N_DIM / 4)           // 512 float4 per row

// ---------------- Stage 1: weighted temporal reduction --------------------
__global__ void __launch_bounds__(256)
spike_decay_stage1(const float* __restrict__ spikes,   // [B, T, N]
                   float* __restrict__ frp)            // [CHUNKS, B, N]
{
    __shared__ float wdecay[TCHUNK];

    const int chunk = blockIdx.y;
    const int t0    = chunk * TCHUNK;

    // decay[t] = exp(-t/10) / sum_t exp(-t/10);  sum = (1-e^-100)/(1-e^-0.1)
    // e^-100 underflows fp32 -> 1/sum == (1 - e^-0.1)
    const float invSum = 1.0f - expf(-0.1f);
    for (int t = threadIdx.x; t < TCHUNK; t += blockDim.x)
        wdecay[t] = expf(-(float)(t0 + t) * 0.1f) * invSum;
    __syncthreads();

    const int flat = blockIdx.x * blockDim.x + threadIdx.x;  // 0 .. B*N4-1
    const int b    = flat >> 9;       // / 512
    const int n4   = flat & 511;      // % 512

    const v4f* __restrict__ p =
        (const v4f*)(spikes + ((size_t)b * T_DIM + t0) * N_DIM) + n4;

    v4f acc = {0.0f, 0.0f, 0.0f, 0.0f};
#pragma unroll 10
    for (int t = 0; t < TCHUNK; ++t) {
        v4f v = __builtin_nontemporal_load(p + (size_t)t * N4);
        const float w = wdecay[t];
        acc.x += v.x * w;
        acc.y += v.y * w;
        acc.z += v.z * w;
        acc.w += v.w * w;
    }
    v4f* out4 = (v4f*)(frp + (size_t)chunk * (B_DIM * N_DIM)) + flat;
    *out4 = acc;
}

// ---------------- Stage 2: WMMA fp32 classifier GEMM ----------------------
// One wave per 16x16 output tile. grid = (4 M-tiles, 3 N-tiles), block = 32.
// A = firing_rates [64 x 2048] (summed over 4 chunk partials)
// B = W^T          [2048 x 48] (cols >= 35 zero-padded via guarded loads)
__global__ void __launch_bounds__(32)
readout_wmma_stage2(const float* __restrict__ frp,   // [CHUNKS, B, N]
                    const float* __restrict__ W,     // [O, N]
                    const float* __restrict__ bias,  // [O]
                    float* __restrict__ out)         // [B, O]
{
    const int m0   = blockIdx.x * 16;
    const int n0   = blockIdx.y * 16;
    const int lane = threadIdx.x;       // 0..31
    const int half = lane >> 4;         // 0: K+0/K+1   1: K+2/K+3
    const int l    = lane & 15;
    const int m    = m0 + l;            // A-fragment row for this lane
    const int n    = n0 + l;            // B-fragment col for this lane
    const int koff = half * 2;

    const bool nvalid = (n < O_DIM);
    const float* __restrict__ Arow = frp + (size_t)m * N_DIM;
    const float* __restrict__ Brow = nvalid ? (W + (size_t)n * N_DIM) : W;

    v8f c = {};
#pragma unroll 2
    for (int k = 0; k < N_DIM; k += 4) {
        const int kb = k + koff;
        // A: reduce the 4 stage-1 partials while building the fragment
        float ax = 0.0f, ay = 0.0f;
#pragma unroll
        for (int ch = 0; ch < CHUNKS; ++ch) {
            const float* p = Arow + (size_t)ch * (B_DIM * N_DIM) + kb;
            ax += p[0];
            ay += p[1];
        }
        v2f a;  a.x = ax;  a.y = ay;
        v2f bf;
        if (nvalid) { bf.x = Brow[kb]; bf.y = Brow[kb + 1]; }
        else        { bf.x = 0.0f;     bf.y = 0.0f;         }

        // D = A(16x4,f32) x B(4x16,f32) + C(16x16,f32)
        c = __builtin_amdgcn_wmma_f32_16x16x4_f32(
                /*neg_a=*/false, a, /*neg_b=*/false, bf,
                /*c_mod=*/(short)0, c, /*reuse_a=*/false, /*reuse_b=*/false);
    }

    // C/D layout: lane -> N = l, VGPR r -> M = r + half*8
    if (nvalid) {
        const float bv = bias[n];
#pragma unroll
        for (int r = 0; r < 8; ++r) {
            const int mrow = m0 + r + half * 8;
            out[(size_t)mrow * O_DIM + n] = c[r] + bv;
        }
    }
}

// ---------------------------------------------------------------------------
extern "C" void kernel_launch(void* const* d_in, const int* in_sizes, int n_in,
                              void* d_out, int out_size, void* d_ws, size_t ws_size,
                              hipStream_t stream) {
    const float* spikes = (const float*)d_in[0];   // [64,1000,2048] f32
    const float* W      = (const float*)d_in[1];   // [35,2048] f32
    const float* bias   = (const float*)d_in[2];   // [35] f32
    float*       out    = (float*)d_out;           // [64,35] f32
    float*       frp    = (float*)d_ws;            // CHUNKS*64*2048 f32 = 2 MB

    // Stage 1: 64*512 threads per chunk, 4 chunks -> 4096 waves streaming HBM
    dim3 g1((B_DIM * N4) / 256, CHUNKS);
    spike_decay_stage1<<<g1, 256, 0, stream>>>(spikes, frp);

    // Stage 2: 12 tiles of 16x16, one wave each
    dim3 g2(B_DIM / 16, (O_DIM + 15) / 16);  // (4, 3)
    readout_wmma_stage2<<<g2, 32, 0, stream>>>(frp, W, bias, out);
}